// VoxelDownsampling_65841848648498
// MI455X (gfx1250) — compile-verified
//
#include <hip/hip_runtime.h>
#include <stdint.h>

typedef uint32_t u32;

#define BLK 256
#define SCAN_IPT 16
#define SCAN_TILE (BLK * SCAN_IPT)
#define I32MAX 0x7fffffff
#define I32MIN (-0x7fffffff - 1)

typedef unsigned int u32x4 __attribute__((ext_vector_type(4)));
typedef int i32x8 __attribute__((ext_vector_type(8)));
typedef int i32x4 __attribute__((ext_vector_type(4)));

#if defined(__gfx1250__) && __has_builtin(__builtin_amdgcn_tensor_load_to_lds) && __has_builtin(__builtin_amdgcn_s_wait_tensorcnt)
#define HAVE_TDM 1
#endif

__device__ __forceinline__ int voxel_coord(float p, float l) {
  return (int)floorf(p / l);  // matches jnp.floor(points/leaf).astype(int)
}

#ifdef HAVE_TDM
// 1-D Tensor Data Mover load: nElems u32 from gptr -> LDS byte offset ldsAddr,
// tile_dim0 = tileElems (OOB reads beyond nElems return 0). Used as an async
// prefetch only -- the committed data path is always regular VMEM loads.
__device__ __forceinline__ void tdm_prefetch_u32(const void* gptr, u32 ldsAddr,
                                                 u32 nElems, u32 tileElems) {
  uint64_t ga = (uint64_t)(uintptr_t)gptr;
  u32x4 g0;
  g0.x = 1u;                                              // count=1, user descriptor
  g0.y = ldsAddr;                                         // lds_addr (bytes)
  g0.z = (uint32_t)ga;                                    // global_addr[31:0]
  g0.w = ((uint32_t)(ga >> 32) & 0x01FFFFFFu) | (2u << 30);  // ga[56:32] | type=2
  i32x8 g1;
  g1[0] = (int)(2u << 16);                                // wg_mask=0, data_size=4B
  g1[1] = (int)((nElems & 0xFFFFu) << 16);                // tensor_dim0[15:0]
  g1[2] = (int)((nElems >> 16) & 0xFFFFu);                // tensor_dim0[31:16], dim1=0
  g1[3] = (int)(tileElems << 16);                         // tile_dim0
  g1[4] = 0;                                              // tile_dim1=0 (1-D)
  g1[5] = (int)tileElems;                                 // tensor_dim0_stride (unused)
  g1[6] = 0;
  g1[7] = 0;
  i32x4 gz4 = {0, 0, 0, 0};
  i32x8 gz8 = {0, 0, 0, 0, 0, 0, 0, 0};
  __builtin_amdgcn_tensor_load_to_lds(g0, g1, gz4, gz4, gz8, 0);
}
#endif

// ---------------------------------------------------------------- utilities
__global__ __launch_bounds__(BLK) void k_zero32(u32* p, long long n) {
  long long i = (long long)blockIdx.x * BLK + threadIdx.x;
  long long stride = (long long)gridDim.x * BLK;
  for (; i < n; i += stride) p[i] = 0u;
}

__global__ void k_init_minmax(int* mm) {
  int t = threadIdx.x;
  if (t < 3) mm[t] = I32MAX;
  else if (t < 6) mm[t] = I32MIN;
}

// ------------------------------------------------- per-axis voxel min / max
__global__ __launch_bounds__(BLK) void k_minmax(const float* __restrict__ pts,
                                                const float* __restrict__ leaf,
                                                int n, int* mm) {
  __shared__ int smin[3][BLK];
  __shared__ int smax[3][BLK];
  float l0 = leaf[0], l1 = leaf[1], l2 = leaf[2];
  int lmin0 = I32MAX, lmin1 = I32MAX, lmin2 = I32MAX;
  int lmax0 = I32MIN, lmax1 = I32MIN, lmax2 = I32MIN;
  long long i = (long long)blockIdx.x * BLK + threadIdx.x;
  long long stride = (long long)gridDim.x * BLK;
  for (; i < n; i += stride) {
    int v0 = voxel_coord(pts[3 * i + 0], l0);
    int v1 = voxel_coord(pts[3 * i + 1], l1);
    int v2 = voxel_coord(pts[3 * i + 2], l2);
    lmin0 = v0 < lmin0 ? v0 : lmin0; lmax0 = v0 > lmax0 ? v0 : lmax0;
    lmin1 = v1 < lmin1 ? v1 : lmin1; lmax1 = v1 > lmax1 ? v1 : lmax1;
    lmin2 = v2 < lmin2 ? v2 : lmin2; lmax2 = v2 > lmax2 ? v2 : lmax2;
  }
  int t = threadIdx.x;
  smin[0][t] = lmin0; smin[1][t] = lmin1; smin[2][t] = lmin2;
  smax[0][t] = lmax0; smax[1][t] = lmax1; smax[2][t] = lmax2;
  __syncthreads();
  for (int st = BLK / 2; st > 0; st >>= 1) {
    if (t < st) {
#pragma unroll
      for (int a = 0; a < 3; ++a) {
        int m0 = smin[a][t + st]; if (m0 < smin[a][t]) smin[a][t] = m0;
        int m1 = smax[a][t + st]; if (m1 > smax[a][t]) smax[a][t] = m1;
      }
    }
    __syncthreads();
  }
  if (t == 0) {
#pragma unroll
    for (int a = 0; a < 3; ++a) {
      atomicMin(&mm[a], smin[a][0]);
      atomicMax(&mm[3 + a], smax[a][0]);
    }
  }
}

// ------------------------------------------------------------- key building
// key = vc0*dim1*dim2 + vc1*dim2 + vc2 (identical to reference); fits u32 here.
__global__ __launch_bounds__(BLK) void k_make_keys(const float* __restrict__ pts,
                                                   const float* __restrict__ leaf,
                                                   int n, const int* __restrict__ mm,
                                                   u32* __restrict__ keys,
                                                   u32* __restrict__ idx) {
#ifdef HAVE_TDM
  __shared__ u32 tdmStage[BLK * 3];
  if (threadIdx.x < 32) {
    long long remain = (long long)n - (long long)blockIdx.x * BLK;
    if (remain > BLK) remain = BLK;
    if (remain < 0) remain = 0;
    tdm_prefetch_u32(pts + (size_t)blockIdx.x * BLK * 3,
                     (u32)(uintptr_t)(void*)&tdmStage[0],
                     (u32)(remain * 3), (u32)(BLK * 3));
  }
#endif
  int i = blockIdx.x * BLK + threadIdx.x;
  if (i < n) {
    float l0 = leaf[0], l1 = leaf[1], l2 = leaf[2];
    int v0 = voxel_coord(pts[3 * (size_t)i + 0], l0) - mm[0];
    int v1 = voxel_coord(pts[3 * (size_t)i + 1], l1) - mm[1];
    int v2 = voxel_coord(pts[3 * (size_t)i + 2], l2) - mm[2];
    uint64_t d1 = (uint64_t)(mm[4] - mm[1] + 1);
    uint64_t d2 = (uint64_t)(mm[5] - mm[2] + 1);
    uint64_t key = (uint64_t)v0 * d1 * d2 + (uint64_t)v1 * d2 + (uint64_t)v2;
    keys[i] = (u32)key;
    idx[i] = (u32)i;
  }
#ifdef HAVE_TDM
  __builtin_amdgcn_s_wait_tensorcnt(0);
#endif
}

// ----------------------------------------------- wave32 ballot digit match
// 9-bit match: digit 0..255 for valid lanes, 256 sentinel for tail lanes so
// EXEC stays full across the ballots and tail lanes never pollute counts.
__device__ __forceinline__ unsigned match_ext(unsigned ext) {
  unsigned m = 0xFFFFFFFFu;
#pragma unroll
  for (int b = 0; b < 9; ++b) {
    unsigned bb = (unsigned)__ballot((int)((ext >> b) & 1u));
    m &= ((ext >> b) & 1u) ? bb : ~bb;
  }
  return m;
}

// ------------------------------------------------------- radix sort kernels
__global__ __launch_bounds__(BLK) void k_radix_hist(const u32* __restrict__ keys,
                                                    int n, int numTiles,
                                                    u32* __restrict__ hist, int shift) {
  __shared__ u32 wh[8][256];
  int tile = blockIdx.x, tid = threadIdx.x;
  int i = tile * BLK + tid;
  int valid = (i < n);
  unsigned ext = valid ? ((keys[i] >> shift) & 0xFFu) : 256u;
  unsigned m = match_ext(ext);
  unsigned lane = (unsigned)tid & 31u;
  unsigned w = (unsigned)tid >> 5;
  unsigned waveRank = __popc(m & ((1u << lane) - 1u));
  unsigned waveCnt = __popc(m);
  for (int j = tid; j < 8 * 256; j += BLK) ((u32*)wh)[j] = 0u;
  __syncthreads();
  if (ext < 256u && waveRank == 0u) wh[w][ext] = waveCnt;
  __syncthreads();
  u32 tot = 0;
#pragma unroll
  for (int ww = 0; ww < 8; ++ww) tot += wh[ww][tid];
  hist[(size_t)tid * (size_t)numTiles + (size_t)tile] = tot;  // digit-major
}

__global__ __launch_bounds__(BLK) void k_radix_scatter(const u32* __restrict__ keysIn,
                                                       const u32* __restrict__ idxIn,
                                                       u32* __restrict__ keysOut,
                                                       u32* __restrict__ idxOut,
                                                       int n, int numTiles,
                                                       const u32* __restrict__ histEx,
                                                       int shift) {
  __shared__ u32 wh[8][256];
  int tile = blockIdx.x, tid = threadIdx.x;
  int i = tile * BLK + tid;
  int valid = (i < n);
  u32 key = valid ? keysIn[i] : 0u;
  u32 pay = valid ? idxIn[i] : 0u;
  unsigned ext = valid ? ((key >> shift) & 0xFFu) : 256u;
  unsigned m = match_ext(ext);
  unsigned lane = (unsigned)tid & 31u;
  unsigned w = (unsigned)tid >> 5;
  unsigned waveRank = __popc(m & ((1u << lane) - 1u));
  unsigned waveCnt = __popc(m);
  for (int j = tid; j < 8 * 256; j += BLK) ((u32*)wh)[j] = 0u;
  __syncthreads();
  if (ext < 256u && waveRank == 0u) wh[w][ext] = waveCnt;
  __syncthreads();
  // exclusive prefix over waves for digit == tid (stable wave-major order)
  u32 run = 0;
#pragma unroll
  for (int ww = 0; ww < 8; ++ww) { u32 tcnt = wh[ww][tid]; wh[ww][tid] = run; run += tcnt; }
  __syncthreads();
  if (valid) {
    u32 base = histEx[(size_t)ext * (size_t)numTiles + (size_t)tile];
    u32 dst = base + wh[w][ext] + waveRank;
    keysOut[dst] = key;
    idxOut[dst] = pay;
  }
}

// -------------------------------------------------------- generic u32 scan
__global__ __launch_bounds__(BLK) void k_scan_block_sums(const u32* __restrict__ d,
                                                         long long L,
                                                         u32* __restrict__ bsums) {
  __shared__ u32 red[BLK];
  long long base = (long long)blockIdx.x * SCAN_TILE;
  u32 s = 0;
#pragma unroll
  for (int k = 0; k < SCAN_IPT; ++k) {
    long long i = base + (long long)k * BLK + threadIdx.x;
    if (i < L) s += d[i];
  }
  red[threadIdx.x] = s;
  __syncthreads();
  for (int st = BLK / 2; st > 0; st >>= 1) {
    if ((int)threadIdx.x < st) red[threadIdx.x] += red[threadIdx.x + st];
    __syncthreads();
  }
  if (threadIdx.x == 0) bsums[blockIdx.x] = red[0];
}

__global__ __launch_bounds__(BLK) void k_scan_spine(u32* bsums, int nb) {
  __shared__ u32 s[2048];
  if (nb <= 2048) {
    for (int i = threadIdx.x; i < nb; i += BLK) s[i] = bsums[i];
    __syncthreads();
    if (threadIdx.x == 0) {
      u32 run = 0;
      for (int i = 0; i < nb; ++i) { u32 t = s[i]; s[i] = run; run += t; }
    }
    __syncthreads();
    for (int i = threadIdx.x; i < nb; i += BLK) bsums[i] = s[i];
  } else {
    if (threadIdx.x == 0) {
      u32 run = 0;
      for (int i = 0; i < nb; ++i) { u32 t = bsums[i]; bsums[i] = run; run += t; }
    }
  }
}

// wave32 shfl scan + one LDS exchange of the 8 wave sums: 2 barriers per round
// instead of 17 for a Hillis-Steele over 256 lanes.
__global__ __launch_bounds__(BLK) void k_scan_apply(u32* __restrict__ d, long long L,
                                                    const u32* __restrict__ bsums,
                                                    int exclusive) {
  __shared__ u32 swsum[8];
  long long base = (long long)blockIdx.x * SCAN_TILE;
  u32 carry = bsums[blockIdx.x];
  int tid = threadIdx.x;
  unsigned lane = (unsigned)tid & 31u;
  unsigned w = (unsigned)tid >> 5;
  for (int k = 0; k < SCAN_IPT; ++k) {
    long long i = base + (long long)k * BLK + tid;
    u32 v = (i < L) ? d[i] : 0u;
    u32 x = v;
#pragma unroll
    for (int off = 1; off < 32; off <<= 1) {
      u32 t = (u32)__shfl_up((int)x, off, 32);
      if (lane >= (unsigned)off) x += t;
    }
    if (lane == 31u) swsum[w] = x;
    __syncthreads();
    u32 waveOff = 0, tot = 0;
#pragma unroll
    for (int ww = 0; ww < 8; ++ww) {
      u32 s = swsum[ww];
      if ((unsigned)ww < w) waveOff += s;
      tot += s;
    }
    __syncthreads();  // swsum fully consumed before next round overwrites it
    u32 incl = x + waveOff + carry;
    u32 res = exclusive ? (incl - v) : incl;
    if (i < L) d[i] = res;
    carry += tot;
  }
}

// ------------------------------------------------ group ids / accumulation
__global__ __launch_bounds__(BLK) void k_flags(const u32* __restrict__ sk, int n,
                                               u32* __restrict__ flags) {
  int i = blockIdx.x * BLK + threadIdx.x;
  if (i < n) flags[i] = (i == 0 || sk[i] != sk[i - 1]) ? 1u : 0u;
}

__global__ __launch_bounds__(BLK) void k_accumulate(const u32* __restrict__ gidIncl,
                                                    const u32* __restrict__ idxSorted,
                                                    const float* __restrict__ pts, int n,
                                                    float* __restrict__ out) {
#ifdef HAVE_TDM
  // Async TDM staging of this block's sorted-index tile (prefetch only).
  __shared__ u32 tdmIdx[BLK];
  if (threadIdx.x < 32) {
    long long remain = (long long)n - (long long)blockIdx.x * BLK;
    if (remain > BLK) remain = BLK;
    if (remain < 0) remain = 0;
    tdm_prefetch_u32(idxSorted + (size_t)blockIdx.x * BLK,
                     (u32)(uintptr_t)(void*)&tdmIdx[0], (u32)remain, (u32)BLK);
  }
#endif
  int i = blockIdx.x * BLK + threadIdx.x;
  if (i < n) {
    u32 j = idxSorted[i];
#if defined(__gfx1250__)
    __builtin_prefetch(&pts[3 * (size_t)j], 0, 0);  // global_prefetch_b8 (gather warm-up)
#endif
    u32 g = gidIncl[i] - 1u;
    float x = pts[3 * (size_t)j + 0];
    float y = pts[3 * (size_t)j + 1];
    float z = pts[3 * (size_t)j + 2];
    atomicAdd(&out[3 * (size_t)g + 0], x);
    atomicAdd(&out[3 * (size_t)g + 1], y);
    atomicAdd(&out[3 * (size_t)g + 2], z);
    atomicAdd(&out[3 * (size_t)n + (size_t)g], 1.0f);  // counts in mask region
  }
#ifdef HAVE_TDM
  __builtin_amdgcn_s_wait_tensorcnt(0);
#endif
}

__global__ __launch_bounds__(BLK) void k_finalize(float* __restrict__ out, int n) {
  int g = blockIdx.x * BLK + threadIdx.x;
  if (g >= n) return;
  float c = out[3 * (size_t)n + (size_t)g];
  if (c > 0.0f) {
    float inv = 1.0f / c;
    out[3 * (size_t)g + 0] *= inv;
    out[3 * (size_t)g + 1] *= inv;
    out[3 * (size_t)g + 2] *= inv;
    out[3 * (size_t)n + (size_t)g] = 1.0f;
  } else {
    out[3 * (size_t)n + (size_t)g] = 0.0f;
  }
}

// --------------------------------------------------------------- launcher
extern "C" void kernel_launch(void* const* d_in, const int* in_sizes, int n_in,
                              void* d_out, int out_size, void* d_ws, size_t ws_size,
                              hipStream_t stream) {
  const float* pts = (const float*)d_in[0];
  const float* leaf = (const float*)d_in[1];
  float* out = (float*)d_out;
  int n = in_sizes[0] / 3;
  int numTiles = (n + BLK - 1) / BLK;
  long long Lh = (long long)numTiles * 256;  // digit-major histogram length

  // workspace layout (u32 elements): keysA | idxA | keysB | idxB | hist | bsums | mm
  u32* ws = (u32*)d_ws;
  size_t nn = (size_t)n;
  u32* keysA = ws;
  u32* idxA = ws + nn;
  u32* keysB = ws + 2 * nn;
  u32* idxB = ws + 3 * nn;
  u32* hist = ws + 4 * nn;              // capacity nn + 512 >= Lh
  u32* bsums = ws + 5 * nn + 512;       // capacity 4608
  int* mm = (int*)(ws + 5 * nn + 512 + 4608);

  k_zero32<<<4096, BLK, 0, stream>>>((u32*)out, 4LL * (long long)n);
  k_init_minmax<<<1, 64, 0, stream>>>(mm);
  k_minmax<<<2048, BLK, 0, stream>>>(pts, leaf, n, mm);
  k_make_keys<<<numTiles, BLK, 0, stream>>>(pts, leaf, n, mm, keysA, idxA);

  u32* curK = keysA; u32* curI = idxA;
  u32* nxtK = keysB; u32* nxtI = idxB;
  int nbH = (int)((Lh + SCAN_TILE - 1) / SCAN_TILE);
  for (int pass = 0; pass < 4; ++pass) {
    int shift = pass * 8;
    k_radix_hist<<<numTiles, BLK, 0, stream>>>(curK, n, numTiles, hist, shift);
    k_scan_block_sums<<<nbH, BLK, 0, stream>>>(hist, Lh, bsums);
    k_scan_spine<<<1, BLK, 0, stream>>>(bsums, nbH);
    k_scan_apply<<<nbH, BLK, 0, stream>>>(hist, Lh, bsums, 1);
    k_radix_scatter<<<numTiles, BLK, 0, stream>>>(curK, curI, nxtK, nxtI, n, numTiles,
                                                  hist, shift);
    u32* tk = curK; curK = nxtK; nxtK = tk;
    u32* ti = curI; curI = nxtI; nxtI = ti;
  }
  // 4 passes: sorted (key, original-index) pairs now in keysA/idxA (== curK/curI)

  k_flags<<<numTiles, BLK, 0, stream>>>(curK, n, hist);  // reuse hist as flags/gid
  int nbF = (int)(((long long)n + SCAN_TILE - 1) / SCAN_TILE);
  k_scan_block_sums<<<nbF, BLK, 0, stream>>>(hist, n, bsums);
  k_scan_spine<<<1, BLK, 0, stream>>>(bsums, nbF);
  k_scan_apply<<<nbF, BLK, 0, stream>>>(hist, n, bsums, 0);  // inclusive -> gid+1

  k_accumulate<<<numTiles, BLK, 0, stream>>>(hist, curI, pts, n, out);
  k_finalize<<<numTiles, BLK, 0, stream>>>(out, n);
}